// GIN_66623532695859
// MI455X (gfx1250) — compile-verified
//
#include <hip/hip_runtime.h>

typedef float v2f __attribute__((ext_vector_type(2)));
typedef float v8f __attribute__((ext_vector_type(8)));

#define N_NODES 50000
#define N_EDGES 800000
#define HID 128
#define LDS_STRIDE 132            // 128 + 4 pad -> conflict-free A-frag reads
#define WAVES_PER_BLOCK 4

// D(16x16,f32) = A(16x4,f32) * B(4x16,f32) + C   — full-precision CDNA5 WMMA
__device__ __forceinline__ v8f wmma_f32(v2f a, v2f b, v8f c) {
  return __builtin_amdgcn_wmma_f32_16x16x4_f32(false, a, false, b, (short)0, c,
                                               false, false);
}

// ---------------------------------------------------------------------------
// Scatter-add: agg[dst[e]] += h[src[e]]   (one wave per edge, 4 floats/lane)
// ---------------------------------------------------------------------------
__global__ __launch_bounds__(256) void gin_scatter(
    const float* __restrict__ h, const int* __restrict__ src,
    const int* __restrict__ dst, float* __restrict__ agg) {
  const int gid  = blockIdx.x * blockDim.x + threadIdx.x;
  const int e    = gid >> 5;
  const int lane = gid & 31;
  if (e >= N_EDGES) return;
  const int s = src[e];
  const int d = dst[e];
  const float4 v = *(const float4*)&h[(size_t)s * HID + (lane << 2)];
  float* out = &agg[(size_t)d * HID + (lane << 2)];
  atomicAdd(out + 0, v.x);
  atomicAdd(out + 1, v.y);
  atomicAdd(out + 2, v.z);
  atomicAdd(out + 3, v.w);
}

// ---------------------------------------------------------------------------
// One GIN layer: h_out = ReLU( ReLU(((1+eps)h + agg) Wa + ba) Wb + bb )
// One wave = one 16-row tile. K=128 via 32 x wmma_f32_16x16x4, 8 N-tiles.
// ---------------------------------------------------------------------------
__global__ __launch_bounds__(128) void gin_layer(
    const float* __restrict__ h_in, const float* __restrict__ agg,
    const float* __restrict__ Wa, const float* __restrict__ ba,
    const float* __restrict__ Wb, const float* __restrict__ bb,
    const float* __restrict__ eps_arr, int layer,
    float* __restrict__ h_out) {
  __shared__ float tile[WAVES_PER_BLOCK][16 * LDS_STRIDE];
  const int wave = threadIdx.x >> 5;
  const int lane = threadIdx.x & 31;
  int row_base = (blockIdx.x * WAVES_PER_BLOCK + wave) * 16;
  if (row_base + 16 > N_NODES) row_base = N_NODES - 16;  // tail waves redo work
  float* T = tile[wave];
  const float epsv = 1.0f + eps_arr[layer];

  // Phase 1: U = (1+eps)*h + agg  ->  LDS, coalesced 512B rows
  for (int i = lane; i < 16 * 32; i += 32) {
    const int r = i >> 5;
    const int c = (i & 31) << 2;
    const size_t off = (size_t)(row_base + r) * HID + c;
    const float4 hv = *(const float4*)&h_in[off];
    const float4 av = *(const float4*)&agg[off];
    float* d = &T[r * LDS_STRIDE + c];
    d[0] = epsv * hv.x + av.x;
    d[1] = epsv * hv.y + av.y;
    d[2] = epsv * hv.z + av.z;
    d[3] = epsv * hv.w + av.w;
  }
  __syncthreads();

  const int m    = lane & 15;            // A-row / B,C-column index
  const int kh   = (lane >> 4) << 1;     // K sub-offset for this half-wave
  const int mrow = (lane >> 4) << 3;     // C/D row base (M split at lane 16)

  const v8f vzero = {0.f, 0.f, 0.f, 0.f, 0.f, 0.f, 0.f, 0.f};
  v8f acc[8];

  // ---- GEMM 1: Y1 = ReLU(U @ Wa + ba) ----
#pragma unroll
  for (int nt = 0; nt < 8; ++nt) acc[nt] = vzero;
  for (int kk = 0; kk < 32; ++kk) {
    const int k0 = (kk << 2) + kh;
    const v2f a = *(const v2f*)&T[m * LDS_STRIDE + k0];
#pragma unroll
    for (int nt = 0; nt < 8; ++nt) {
      const float* wp = Wa + (size_t)k0 * HID + nt * 16 + m;
      v2f b;
      b.x = wp[0];
      b.y = wp[HID];
      acc[nt] = wmma_f32(a, b, acc[nt]);
    }
  }
  __syncthreads();  // all waves done reading U
#pragma unroll
  for (int nt = 0; nt < 8; ++nt) {
    const int col = nt * 16 + m;
    const float bias = ba[col];
#pragma unroll
    for (int i = 0; i < 8; ++i) {
      const float v = acc[nt][i] + bias;
      T[(mrow + i) * LDS_STRIDE + col] = v > 0.0f ? v : 0.0f;
    }
  }
  __syncthreads();

  // ---- GEMM 2: H = ReLU(Y1 @ Wb + bb) -> global ----
#pragma unroll
  for (int nt = 0; nt < 8; ++nt) acc[nt] = vzero;
  for (int kk = 0; kk < 32; ++kk) {
    const int k0 = (kk << 2) + kh;
    const v2f a = *(const v2f*)&T[m * LDS_STRIDE + k0];
#pragma unroll
    for (int nt = 0; nt < 8; ++nt) {
      const float* wp = Wb + (size_t)k0 * HID + nt * 16 + m;
      v2f b;
      b.x = wp[0];
      b.y = wp[HID];
      acc[nt] = wmma_f32(a, b, acc[nt]);
    }
  }
#pragma unroll
  for (int nt = 0; nt < 8; ++nt) {
    const int col = nt * 16 + m;
    const float bias = bb[col];
#pragma unroll
    for (int i = 0; i < 8; ++i) {
      const float v = acc[nt][i] + bias;
      h_out[(size_t)(row_base + mrow + i) * HID + col] = v > 0.0f ? v : 0.0f;
    }
  }
}

// ---------------------------------------------------------------------------
// Head: out = ReLU(h @ Wh1 + bh1) @ Wh2 + bh2   (128 -> 64 -> 2)
// ---------------------------------------------------------------------------
__global__ __launch_bounds__(128) void gin_head(
    const float* __restrict__ h, const float* __restrict__ Wh1,
    const float* __restrict__ bh1, const float* __restrict__ Wh2,
    const float* __restrict__ bh2, float* __restrict__ out) {
  __shared__ float tile[WAVES_PER_BLOCK][16 * LDS_STRIDE];
  const int wave = threadIdx.x >> 5;
  const int lane = threadIdx.x & 31;
  int row_base = (blockIdx.x * WAVES_PER_BLOCK + wave) * 16;
  if (row_base + 16 > N_NODES) row_base = N_NODES - 16;
  float* T = tile[wave];

  for (int i = lane; i < 16 * 32; i += 32) {
    const int r = i >> 5;
    const int c = (i & 31) << 2;
    *(float4*)&T[r * LDS_STRIDE + c] =
        *(const float4*)&h[(size_t)(row_base + r) * HID + c];
  }
  __syncthreads();

  const int m    = lane & 15;
  const int kh   = (lane >> 4) << 1;
  const int mrow = (lane >> 4) << 3;

  const v8f vzero = {0.f, 0.f, 0.f, 0.f, 0.f, 0.f, 0.f, 0.f};
  v8f acc[4];
#pragma unroll
  for (int nt = 0; nt < 4; ++nt) acc[nt] = vzero;
  for (int kk = 0; kk < 32; ++kk) {
    const int k0 = (kk << 2) + kh;
    const v2f a = *(const v2f*)&T[m * LDS_STRIDE + k0];
#pragma unroll
    for (int nt = 0; nt < 4; ++nt) {
      const float* wp = Wh1 + (size_t)k0 * 64 + nt * 16 + m;
      v2f b;
      b.x = wp[0];
      b.y = wp[64];
      acc[nt] = wmma_f32(a, b, acc[nt]);
    }
  }
  __syncthreads();  // done reading input tile
#pragma unroll
  for (int nt = 0; nt < 4; ++nt) {
    const int col = nt * 16 + m;
    const float bias = bh1[col];
#pragma unroll
    for (int i = 0; i < 8; ++i) {
      const float v = acc[nt][i] + bias;
      T[(mrow + i) * LDS_STRIDE + col] = v > 0.0f ? v : 0.0f;  // hid tile 16x64
    }
  }
  __syncthreads();

  // 64 -> 2: lane -> (row = lane>>1, j = lane&1)
  const int rm = lane >> 1;
  const int j  = lane & 1;
  float sum = bh2[j];
  for (int k = 0; k < 64; ++k) sum += T[rm * LDS_STRIDE + k] * Wh2[k * 2 + j];
  out[(size_t)(row_base + rm) * 2 + j] = sum;
}

// ---------------------------------------------------------------------------
extern "C" void kernel_launch(void* const* d_in, const int* in_sizes, int n_in,
                              void* d_out, int out_size, void* d_ws,
                              size_t ws_size, hipStream_t stream) {
  (void)in_sizes; (void)n_in; (void)out_size; (void)ws_size;
  const float* x   = (const float*)d_in[0];
  const int*   ei  = (const int*)d_in[1];   // [2, N_EDGES]
  const float* eps = (const float*)d_in[2];
  const float* Wa  = (const float*)d_in[3];  // [3,128,128]
  const float* ba  = (const float*)d_in[4];  // [3,128]
  const float* Wb  = (const float*)d_in[5];
  const float* bb  = (const float*)d_in[6];
  const float* Wh1 = (const float*)d_in[7];  // [128,64]
  const float* bh1 = (const float*)d_in[8];
  const float* Wh2 = (const float*)d_in[9];  // [64,2]
  const float* bh2 = (const float*)d_in[10];
  const int* src = ei;
  const int* dst = ei + N_EDGES;

  const size_t feat = (size_t)N_NODES * HID;
  float* h0  = (float*)d_ws;
  float* h1  = h0 + feat;
  float* agg = h1 + feat;                   // needs 3*25.6 MB of d_ws

  const int scatter_blocks = (N_EDGES * 32) / 256;                       // 100000
  const int gemm_blocks = (N_NODES / 16 + WAVES_PER_BLOCK - 1) / WAVES_PER_BLOCK;

  const float* cur = x;
  float* nxt[3] = {h0, h1, h0};
  for (int l = 0; l < 3; ++l) {
    hipMemsetAsync(agg, 0, feat * sizeof(float), stream);
    gin_scatter<<<scatter_blocks, 256, 0, stream>>>(cur, src, dst, agg);
    gin_layer<<<gemm_blocks, 128, 0, stream>>>(
        cur, agg, Wa + (size_t)l * HID * HID, ba + l * HID,
        Wb + (size_t)l * HID * HID, bb + l * HID, eps, l, nxt[l]);
    cur = nxt[l];
  }
  gin_head<<<gemm_blocks, 128, 0, stream>>>(cur, Wh1, bh1, Wh2, bh2,
                                            (float*)d_out);
}